// DeviceQCP_30803505447120
// MI455X (gfx1250) — compile-verified
//
#include <hip/hip_runtime.h>

// ============================================================================
// QP sensitivity solve (CG on normal equations) for MI455X / gfx1250.
// Memory-bound sparse pipeline: atomic-scatter SpMV (B128 streaming loads) +
// fused elementwise + WMMA wave32 dot reductions (v_wmma_f32_16x16x4_f32).
// All CG scalars live in device workspace -> fully graph-capturable, no sync.
// ============================================================================

#define TPB 256
#define DOTGRID 240

typedef __attribute__((ext_vector_type(2))) float v2f;
typedef __attribute__((ext_vector_type(8))) float v8f;

// workspace scalar slots
enum { S_XTPX = 0, S_GAMMA, S_GAMNEW, S_PAP, S_ALPHA, S_PCOEF, S_RCOEF,
       S_FROZEN, S_ACC, S_DDT, S_TOL2, S_NSLOTS = 16 };

// ---------------------------------------------------------------------------
// Wave32 all-lanes sum via the matrix pipe:
//   A(16x4): lane L(0..15) holds row L K={0,1}; lane L+16 holds row L K={2,3}.
//   With A = {partial, 0} per lane and B = ones, D[i][j] = partial_i + partial_{i+16}.
//   D layout: VGPR v, lanes0-15 -> M=v, lanes16-31 -> M=v+8, so summing the 8
//   accumulator regs gives half-sums; one xor-16 shuffle completes the wave sum.
// EXEC must be all ones at the call site (no divergence).
// ---------------------------------------------------------------------------
__device__ __forceinline__ float wave_allsum_wmma(float v) {
  v2f a = { v, 0.0f };
  v2f ones = { 1.0f, 1.0f };
  v8f c = {};
  v8f d = __builtin_amdgcn_wmma_f32_16x16x4_f32(
      /*neg_a=*/false, a, /*neg_b=*/false, ones,
      /*c_mod=*/(short)0, c, /*reuse_a=*/false, /*reuse_b=*/false);
  float s = 0.0f;
#pragma unroll
  for (int i = 0; i < 8; ++i) s += d[i];
  s += __shfl_xor(s, 16, 32);   // combine rowsums 0-7 with rowsums 8-15
  return s;                      // every lane now has the full 32-lane sum
}

__device__ __forceinline__ void block_atomic_add(float v, float* target) {
  __shared__ float sw[TPB / 32];
  float ws = wave_allsum_wmma(v);
  int lane = threadIdx.x & 31;
  int wid  = threadIdx.x >> 5;
  if (lane == 0) sw[wid] = ws;
  __syncthreads();
  if (threadIdx.x == 0) {
    float t = 0.0f;
#pragma unroll
    for (int i = 0; i < TPB / 32; ++i) t += sw[i];
    atomicAdd(target, t);   // global_atomic_add_f32, one per block
  }
}

// ---------------------------------------------------------------------------
// Kernels
// ---------------------------------------------------------------------------
__global__ void k_zero(float* p, int len) {
  int i = blockIdx.x * blockDim.x + threadIdx.x;
  if (i < len) p[i] = 0.0f;
}

__global__ void k_copy(float* dst, const float* src, int len) {
  int i = blockIdx.x * blockDim.x + threadIdx.x;
  if (i < len) dst[i] = src[i];
}

__global__ void k_addv(float* dst, const float* a, const float* b, int len) {
  int i = blockIdx.x * blockDim.x + threadIdx.x;
  if (i < len) dst[i] = a[i] + b[i];
}

// mask = (y-s > 0), pi_m = max(y-s, 0)
__global__ void k_setup_masks(const float* y, const float* s, float* mask,
                              float* pim, int m) {
  int i = blockIdx.x * blockDim.x + threadIdx.x;
  if (i < m) {
    float v = y[i] - s[i];
    mask[i] = (v > 0.0f) ? 1.0f : 0.0f;
    pim[i]  = fmaxf(v, 0.0f);
  }
}

__global__ void k_c3(float* c3, const float* q, const float* Px, int n) {
  int i = blockIdx.x * blockDim.x + threadIdx.x;
  if (i < n) c3[i] = fmaf(2.0f, Px[i], q[i]);
}

__global__ void k_mask_mul(float* dst, const float* mask, const float* v, int m) {
  int i = blockIdx.x * blockDim.x + threadIdx.x;
  if (i < m) dst[i] = mask[i] * v[i];
}

// out[rows[e]] += alpha * vals[e] * xin[cols[e]]  (atomic scatter SpMV).
// Streaming side vectorized: 4 nnz/thread via B128 loads of rows/cols/vals
// (all device buffers are hipMalloc-aligned), scalar tail after.
__global__ void k_spmv(const int* __restrict__ rows, const int* __restrict__ cols,
                       const float* __restrict__ vals, const float* __restrict__ xin,
                       float* out, int nnz, float alpha) {
  int tid    = blockIdx.x * blockDim.x + threadIdx.x;
  int stride = gridDim.x * blockDim.x;
  int nq = nnz >> 2;
  const int4*   r4 = (const int4*)rows;
  const int4*   c4 = (const int4*)cols;
  const float4* v4 = (const float4*)vals;
  for (int e = tid; e < nq; e += stride) {
    if (((threadIdx.x & 15) == 0) && (e + stride < nq)) {
      __builtin_prefetch(r4 + e + stride, 0, 1);   // global_prefetch_b8
      __builtin_prefetch(c4 + e + stride, 0, 1);
      __builtin_prefetch(v4 + e + stride, 0, 1);
    }
    int4   rr = r4[e];           // global_load_b128
    int4   cc = c4[e];
    float4 vv = v4[e];
    atomicAdd(out + rr.x, alpha * vv.x * xin[cc.x]);
    atomicAdd(out + rr.y, alpha * vv.y * xin[cc.y]);
    atomicAdd(out + rr.z, alpha * vv.z * xin[cc.z]);
    atomicAdd(out + rr.w, alpha * vv.w * xin[cc.w]);
  }
  for (int e = (nq << 2) + tid; e < nnz; e += stride)
    atomicAdd(out + rows[e], alpha * vals[e] * xin[cols[e]]);
}

// target += alpha * sum(a[i]*b[i])   (B128 loads, WMMA wave reduce, 1 atomic/block)
__global__ void k_dot(const float* __restrict__ a, const float* __restrict__ b,
                      int len, float alpha, float* target) {
  int tid    = blockIdx.x * blockDim.x + threadIdx.x;
  int stride = gridDim.x * blockDim.x;
  int nq = len >> 2;
  const float4* a4 = (const float4*)a;
  const float4* b4 = (const float4*)b;
  float acc = 0.0f;
  for (int i = tid; i < nq; i += stride) {
    float4 xa = a4[i];
    float4 xb = b4[i];
    acc = fmaf(xa.x, xb.x, acc);
    acc = fmaf(xa.y, xb.y, acc);
    acc = fmaf(xa.z, xb.z, acc);
    acc = fmaf(xa.w, xb.w, acc);
  }
  for (int i = (nq << 2) + tid; i < len; i += stride)
    acc = fmaf(a[i], b[i], acc);
  // all lanes reconverged here: EXEC all-1s for the WMMA reduction
  block_atomic_add(alpha * acc, target);
}

// ---- F(v) pieces: out_n = P vn + A^T(mask*vm) + q*vt ; out_m = -A vn + b*vt + (1-mask)*vm
__global__ void k_f_init_n(float* out, const float* q, const float* v, int n, int K) {
  int i = blockIdx.x * blockDim.x + threadIdx.x;
  float vt = v[K - 1];
  if (i < n) out[i] = q[i] * vt;
}

__global__ void k_f_init_m(float* out, const float* b, const float* mask,
                           const float* v, int n, int m, int K) {
  int i = blockIdx.x * blockDim.x + threadIdx.x;
  float vt = v[K - 1];
  if (i < m) out[n + i] = b[i] * vt + (1.0f - mask[i]) * v[n + i];
}

__global__ void k_f_t(float* out, const float* acc, const float* xtpx,
                      const float* v, int K) {
  // out_t = (-c3.vn - b.um) + xTPx*vt  (acc already holds the two dots with -1)
  out[K - 1] = *acc + (*xtpx) * v[K - 1];
}

// ---- FT(sgn*w) pieces: out_n = sgn*(P wn - A^T wm - c3*wt)
__global__ void k_ft_init_n(float* out, const float* c3, const float* w,
                            int n, int K, float sgn) {
  int i = blockIdx.x * blockDim.x + threadIdx.x;
  float wt = w[K - 1];
  if (i < n) out[i] = -sgn * c3[i] * wt;
}

__global__ void k_ft_init_tm(float* tm, const float* b, const float* w,
                             int m, int K, float sgn) {
  int i = blockIdx.x * blockDim.x + threadIdx.x;
  float wt = w[K - 1];
  if (i < m) tm[i] = -sgn * b[i] * wt;
}

__global__ void k_ft_combine_m(float* out, const float* tm, const float* mask,
                               const float* w, int n, int m, float sgn) {
  int i = blockIdx.x * blockDim.x + threadIdx.x;
  if (i < m) out[n + i] = mask[i] * tm[i] + (1.0f - mask[i]) * sgn * w[n + i];
}

__global__ void k_ft_t(float* out, const float* acc, const float* xtpx,
                       const float* w, int K, float sgn) {
  out[K - 1] = sgn * (*acc + (*xtpx) * w[K - 1]);
}

// ---- CG scalar kernels (device-resident control, emulating jax cg tol stop)
__global__ void k_cg_prep(float* scal) {
  float g = scal[S_GAMMA];
  scal[S_TOL2]   = 1e-16f * g;          // (tol=1e-8)^2 * ||rhs||^2
  scal[S_FROZEN] = (g > 0.0f) ? 0.0f : 1.0f;
}

__global__ void k_alpha(float* scal) {
  float g = scal[S_GAMMA], pap = scal[S_PAP];
  float frozen = scal[S_FROZEN];
  if (g <= scal[S_TOL2] || !(pap > 0.0f)) frozen = 1.0f;  // converged/breakdown
  scal[S_FROZEN] = frozen;
  scal[S_ALPHA]  = (frozen != 0.0f) ? 0.0f : g / pap;
}

__global__ void k_beta(float* scal) {
  if (scal[S_FROZEN] != 0.0f) {
    scal[S_PCOEF] = 1.0f;   // p <- p (frozen)
    scal[S_RCOEF] = 0.0f;
  } else {
    float g = scal[S_GAMMA], gn = scal[S_GAMNEW];
    scal[S_PCOEF] = (g > 0.0f) ? gn / g : 0.0f;
    scal[S_RCOEF] = 1.0f;
    scal[S_GAMMA] = gn;
  }
}

__global__ void k_update_zr(float* z, float* r, const float* p, const float* Nv,
                            const float* scal, int K) {
  int i = blockIdx.x * blockDim.x + threadIdx.x;
  float a = scal[S_ALPHA];
  if (i < K) {
    z[i] = fmaf(a, p[i], z[i]);
    r[i] = fmaf(-a, Nv[i], r[i]);
  }
}

__global__ void k_update_p(float* p, const float* r, const float* scal, int K) {
  int i = blockIdx.x * blockDim.x + threadIdx.x;
  float pc = scal[S_PCOEF], rc = scal[S_RCOEF];
  if (i < K) p[i] = fmaf(pc, p[i], rc * r[i]);
}

// dx = dzn - x*dzt ; t = mask*dzm ; dy = t - y*dzt ; ds = t - dzm - s*dzt
__global__ void k_finalize(float* out, const float* z, const float* x,
                           const float* y, const float* s, const float* mask,
                           int n, int m, int K) {
  int i = blockIdx.x * blockDim.x + threadIdx.x;
  float zt = z[K - 1];
  if (i < n) out[i] = z[i] - x[i] * zt;
  if (i < m) {
    float dzm = z[n + i];
    float t = mask[i] * dzm;
    out[n + i]     = t - y[i] * zt;
    out[n + m + i] = t - dzm - s[i] * zt;
  }
}

// ---------------------------------------------------------------------------
extern "C" void kernel_launch(void* const* d_in, const int* in_sizes, int n_in,
                              void* d_out, int out_size, void* d_ws, size_t ws_size,
                              hipStream_t stream) {
  (void)n_in; (void)out_size; (void)ws_size;
  const int*   P_rows  = (const int*)  d_in[0];
  const int*   P_cols  = (const int*)  d_in[1];
  const float* P_vals  = (const float*)d_in[2];
  const int*   A_rows  = (const int*)  d_in[3];
  const int*   A_cols  = (const int*)  d_in[4];
  const float* A_vals  = (const float*)d_in[5];
  const float* q       = (const float*)d_in[6];
  const float* b       = (const float*)d_in[7];
  const float* x       = (const float*)d_in[8];
  const float* y       = (const float*)d_in[9];
  const float* s       = (const float*)d_in[10];
  const float* dP_vals = (const float*)d_in[11];
  const float* dA_vals = (const float*)d_in[12];
  const float* dq      = (const float*)d_in[13];
  const float* db      = (const float*)d_in[14];

  const int nnzP = in_sizes[0];
  const int nnzA = in_sizes[3];
  const int n    = in_sizes[6];
  const int m    = in_sizes[7];
  const int K    = n + m + 1;

  // workspace carve-out (floats); ~11 MB total
  float* W    = (float*)d_ws;
  float* mask = W; W += m;
  float* pim  = W; W += m;
  float* tm   = W; W += m;   // scratch m-vector (um / t_m)
  float* Px   = W; W += n;
  float* dPx  = W; W += n;
  float* c3   = W; W += n;
  float* dd   = W; W += K;
  float* r    = W; W += K;
  float* p    = W; W += K;
  float* w    = W; W += K;
  float* Nv   = W; W += K;
  float* z    = W; W += K;
  float* scal = W; W += S_NSLOTS;

  auto blocks = [](int len) { return dim3((unsigned)((len + TPB - 1) / TPB)); };
  auto zero = [&](float* ptr, int len) {
    k_zero<<<blocks(len), TPB, 0, stream>>>(ptr, len);
  };
  auto dot = [&](const float* a, const float* bb, int len, float alpha, float* tgt) {
    k_dot<<<dim3(DOTGRID), TPB, 0, stream>>>(a, bb, len, alpha, tgt);
  };
  auto spmv = [&](const int* rr, const int* cc, const float* vv,
                  const float* xin, float* out, int nnz, float alpha) {
    int nq = nnz >> 2;                        // quads per thread-iteration
    int g = (nq + TPB - 1) / TPB;
    if (g < 1) g = 1;
    if (g > 16384) g = 16384;
    k_spmv<<<dim3(g), TPB, 0, stream>>>(rr, cc, vv, xin, out, nnz, alpha);
  };

  // F(v): out_n = P vn + A^T(mask*vm) + q*vt ; out_m = -A vn + b*vt + (1-mask)*vm
  //        out_t = -c3.vn - b.(mask*vm) + xTPx*vt
  auto fmv = [&](const float* v, float* out) {
    k_mask_mul<<<blocks(m), TPB, 0, stream>>>(tm, mask, v + n, m);       // um
    k_f_init_n<<<blocks(n), TPB, 0, stream>>>(out, q, v, n, K);
    k_f_init_m<<<blocks(m), TPB, 0, stream>>>(out, b, mask, v, n, m, K);
    spmv(P_rows, P_cols, P_vals, v,  out,     nnzP,  1.0f);              // P vn
    spmv(A_cols, A_rows, A_vals, tm, out,     nnzA,  1.0f);              // A^T um
    spmv(A_rows, A_cols, A_vals, v,  out + n, nnzA, -1.0f);              // -A vn
    zero(scal + S_ACC, 1);
    dot(c3, v,  n, -1.0f, scal + S_ACC);
    dot(b,  tm, m, -1.0f, scal + S_ACC);
    k_f_t<<<dim3(1), 1, 0, stream>>>(out, scal + S_ACC, scal + S_XTPX, v, K);
  };

  // FT(sgn*w): out_n = sgn*(P wn - A^T wm - c3*wt)
  //            out_m = mask*sgn*(A wn - b*wt) + (1-mask)*sgn*wm
  //            out_t = sgn*(q.wn + b.wm + xTPx*wt)
  auto ftmv = [&](const float* win, float* out, float sgn) {
    k_ft_init_n<<<blocks(n), TPB, 0, stream>>>(out, c3, win, n, K, sgn);
    k_ft_init_tm<<<blocks(m), TPB, 0, stream>>>(tm, b, win, m, K, sgn);
    spmv(P_rows, P_cols, P_vals, win,     out, nnzP,  sgn);              // P wn
    spmv(A_cols, A_rows, A_vals, win + n, out, nnzA, -sgn);              // -A^T wm
    spmv(A_rows, A_cols, A_vals, win,     tm,  nnzA,  sgn);              // A wn
    k_ft_combine_m<<<blocks(m), TPB, 0, stream>>>(out, tm, mask, win, n, m, sgn);
    zero(scal + S_ACC, 1);
    dot(q, win,     n, 1.0f, scal + S_ACC);
    dot(b, win + n, m, 1.0f, scal + S_ACC);
    k_ft_t<<<dim3(1), 1, 0, stream>>>(out, scal + S_ACC, scal + S_XTPX, win, K, sgn);
  };

  // ---- setup: masks, Px, xTPx, c3, dPx ----
  k_setup_masks<<<blocks(m), TPB, 0, stream>>>(y, s, mask, pim, m);
  zero(Px, n);
  spmv(P_rows, P_cols, P_vals, x, Px, nnzP, 1.0f);
  zero(scal, S_NSLOTS);
  dot(x, Px, n, 1.0f, scal + S_XTPX);
  k_c3<<<blocks(n), TPB, 0, stream>>>(c3, q, Px, n);
  zero(dPx, n);
  spmv(P_rows, P_cols, dP_vals, x, dPx, nnzP, 1.0f);

  // ---- dd = [dPx + dA^T pi + dq ; -dA x + db ; -dq.x - db.pi - x.dPx] ----
  k_addv<<<blocks(n), TPB, 0, stream>>>(dd, dPx, dq, n);
  k_copy<<<blocks(m), TPB, 0, stream>>>(dd + n, db, m);
  spmv(A_cols, A_rows, dA_vals, pim, dd,     nnzA,  1.0f);
  spmv(A_rows, A_cols, dA_vals, x,   dd + n, nnzA, -1.0f);
  dot(dq, x,   n, -1.0f, scal + S_DDT);
  dot(db, pim, m, -1.0f, scal + S_DDT);
  dot(x,  dPx, n, -1.0f, scal + S_DDT);
  k_copy<<<dim3(1), 1, 0, stream>>>(dd + (K - 1), scal + S_DDT, 1);

  // ---- CG init: rhs = FT(-dd) -> r ; p = r ; z = 0 ; gamma = r.r ----
  ftmv(dd, r, -1.0f);
  k_copy<<<blocks(K), TPB, 0, stream>>>(p, r, K);
  zero(z, K);
  dot(r, r, K, 1.0f, scal + S_GAMMA);
  k_cg_prep<<<dim3(1), 1, 0, stream>>>(scal);

  // ---- 50 CG iterations on FT(F(.)), fully device-side ----
  for (int it = 0; it < 50; ++it) {
    fmv(p, w);
    ftmv(w, Nv, 1.0f);
    zero(scal + S_PAP, 1);
    dot(p, Nv, K, 1.0f, scal + S_PAP);
    k_alpha<<<dim3(1), 1, 0, stream>>>(scal);
    k_update_zr<<<blocks(K), TPB, 0, stream>>>(z, r, p, Nv, scal, K);
    zero(scal + S_GAMNEW, 1);
    dot(r, r, K, 1.0f, scal + S_GAMNEW);
    k_beta<<<dim3(1), 1, 0, stream>>>(scal);
    k_update_p<<<blocks(K), TPB, 0, stream>>>(p, r, scal, K);
  }

  // ---- outputs ----
  k_finalize<<<blocks(m > n ? m : n), TPB, 0, stream>>>(
      (float*)d_out, z, x, y, s, mask, n, m, K);
}